// DynamicGNN_70420283785533
// MI455X (gfx1250) — compile-verified
//
#include <hip/hip_runtime.h>
#include <hip/hip_bf16.h>

#define T_ 32
#define N_ 5000
#define E_ 80000
#define H_ 128
#define TN_ (T_ * N_)           // 160000
#define TNH_ ((size_t)TN_ * H_) // 20,480,000 floats

typedef __attribute__((ext_vector_type(2))) float v2f;
typedef __attribute__((ext_vector_type(8))) float v8f;

// ---------------- degree / normalization ----------------
__global__ void fill_deg(float* deg) {
    int i = blockIdx.x * blockDim.x + threadIdx.x;
    if (i < TN_) deg[i] = 1.0f;  // self loop
}

__global__ void accum_deg(const int* __restrict__ ei, float* __restrict__ deg) {
    int i = blockIdx.x * blockDim.x + threadIdx.x;
    if (i >= T_ * E_) return;
    int t = i / E_, e = i - t * E_;
    int d = ei[(size_t)t * 2 * E_ + E_ + e];
    unsafeAtomicAdd(&deg[t * N_ + d], 1.0f);
}

__global__ void finalize_dinv(float* deg) {
    int i = blockIdx.x * blockDim.x + threadIdx.x;
    if (i < TN_) deg[i] = rsqrtf(deg[i]);  // deg >= 1 always
}

// ------- dense GEMM [M x 128] @ [128 x 128] (fp32 WMMA) + fused self-loop init -------
// Writes Out = h @ W (scatter source) and OutAcc = dinv^2 * (h @ W) (aggregation target,
// pre-seeded with the self-loop contribution so no zero-fill pass is needed).
__global__ __launch_bounds__(256) void gemm_wmma(const float* __restrict__ A,
                                                 const float* __restrict__ W,
                                                 const float* __restrict__ dinv,
                                                 float* __restrict__ Out,
                                                 float* __restrict__ OutAcc) {
    __shared__ float sW[128 * 128];  // 64 KB: full weight matrix
    for (int i = threadIdx.x; i < 128 * 128; i += 256) sW[i] = W[i];
    __syncthreads();

    int wave = threadIdx.x >> 5;
    int lane = threadIdx.x & 31;
    int lrow = lane & 15;
    int hi   = lane >> 4;          // 0 or 1
    int khalf = hi * 2;            // 0 or 2
    int mbase = blockIdx.x * 128 + wave * 16;

    v8f acc[8] = {};
    const float* arow = A + (size_t)(mbase + lrow) * 128;

    for (int k4 = 0; k4 < 32; ++k4) {
        int k = k4 * 4 + khalf;
        v2f a;
        a.x = arow[k];
        a.y = arow[k + 1];
#pragma unroll
        for (int nt = 0; nt < 8; ++nt) {
            int col = nt * 16 + lrow;
            v2f b;
            b.x = sW[k * 128 + col];
            b.y = sW[(k + 1) * 128 + col];
            acc[nt] = __builtin_amdgcn_wmma_f32_16x16x4_f32(
                false, a, false, b, (short)0, acc[nt], false, false);
        }
    }
#pragma unroll
    for (int r = 0; r < 8; ++r) {
        int row = mbase + r + hi * 8;
        float w = dinv[row];
        w *= w;  // self-loop norm dinv^2
#pragma unroll
        for (int nt = 0; nt < 8; ++nt) {
            int col = nt * 16 + lrow;
            float v = acc[nt][r];
            Out[(size_t)row * 128 + col] = v;
            OutAcc[(size_t)row * 128 + col] = v * w;
        }
    }
}

// ---------------- edge scatter: one wave per edge, lane owns 4 consecutive feats ----------------
__global__ void edge_scatter(const float* __restrict__ hW, const int* __restrict__ ei,
                             const float* __restrict__ dinv, float* __restrict__ out) {
    int gwave = (blockIdx.x * blockDim.x + threadIdx.x) >> 5;
    int lane  = threadIdx.x & 31;
    int nwaves = (gridDim.x * blockDim.x) >> 5;
    const long total = (long)T_ * E_;
    for (long idx = gwave; idx < total; idx += nwaves) {
        int t = (int)(idx / E_), e = (int)(idx - (long)t * E_);
        const int* eit = ei + (size_t)t * 2 * E_;
        int s = eit[e], d = eit[E_ + e];
        float w = dinv[t * N_ + s] * dinv[t * N_ + d];
        const float4* src = (const float4*)(hW + ((size_t)t * N_ + s) * 128);
        float4 v = src[lane];                       // one b128 load per lane
        float* dst = out + ((size_t)t * N_ + d) * 128 + lane * 4;
        unsafeAtomicAdd(&dst[0], v.x * w);
        unsafeAtomicAdd(&dst[1], v.y * w);
        unsafeAtomicAdd(&dst[2], v.z * w);
        unsafeAtomicAdd(&dst[3], v.w * w);
    }
}

// ---------------- epilogue: h = tanh(agg + b), float4 ----------------
__global__ void bias_tanh(float* __restrict__ h, const float* __restrict__ b) {
    int i = blockIdx.x * blockDim.x + threadIdx.x;  // over TN_*32 float4 groups
    if (i >= TN_ * 32) return;
    int g = i & 31;
    float4 bv = ((const float4*)b)[g];
    float4 v = ((float4*)h)[i];
    v.x = tanhf(v.x + bv.x);
    v.y = tanhf(v.y + bv.y);
    v.z = tanhf(v.z + bv.z);
    v.w = tanhf(v.w + bv.w);
    ((float4*)h)[i] = v;
}

// ---------------- per-graph sum over nodes -> z[T,128] ----------------
__global__ void reduce_z(const float* __restrict__ h, float* __restrict__ z) {
    int t = blockIdx.x, f = threadIdx.x;  // 128 threads
    const float* base = h + (size_t)t * N_ * 128 + f;
    float s = 0.f;
    for (int n = 0; n < N_; ++n) s += base[(size_t)n * 128];
    z[t * 128 + f] = s;
}

// ---------------- transformer head, single 128-thread block ----------------
__global__ __launch_bounds__(128) void head_kernel(
    const float* __restrict__ z,
    const float* __restrict__ Wq, const float* __restrict__ bq,
    const float* __restrict__ Wk, const float* __restrict__ bk,
    const float* __restrict__ Wv, const float* __restrict__ bv,
    const float* __restrict__ Wo, const float* __restrict__ bo,
    const float* __restrict__ g2, const float* __restrict__ beta2,
    const float* __restrict__ Wm1, const float* __restrict__ bm1,
    const float* __restrict__ Wm2, const float* __restrict__ bm2,
    const float* __restrict__ Wl, const float* __restrict__ bl,
    float* __restrict__ out) {
    __shared__ float sZ[32 * 128];  // z, later o
    __shared__ float sQ[32 * 128];  // q, later x_attend
    __shared__ float sK[32 * 128];  // k, later mlp hidden chunk / pooled
    __shared__ float sV[32 * 128];  // v
    int tid = threadIdx.x;

    for (int i = tid; i < 32 * 128; i += 128) sZ[i] = z[i];
    __syncthreads();

    // q, k, v projections: thread owns column j = tid
    for (int t = 0; t < 32; ++t) {
        float aq = bq[tid], ak = bk[tid], av = bv[tid];
        for (int f = 0; f < 128; ++f) {
            float zv = sZ[t * 128 + f];
            aq += zv * Wq[f * 128 + tid];
            ak += zv * Wk[f * 128 + tid];
            av += zv * Wv[f * 128 + tid];
        }
        sQ[t * 128 + tid] = aq;
        sK[t * 128 + tid] = ak;
        sV[t * 128 + tid] = av;
    }
    __syncthreads();

    // attention: 256 (t,head) pairs, 2 per thread; DH=16, scale 1/sqrt(16)=0.25
    float oacc[2][16];
    for (int pp = 0; pp < 2; ++pp) {
        int p = tid + pp * 128;
        int t = p >> 3, hh = p & 7;
        float sc[32], mx = -1e30f;
#pragma unroll
        for (int s = 0; s < 32; ++s) {
            float dot = 0.f;
#pragma unroll
            for (int d = 0; d < 16; ++d)
                dot += sQ[t * 128 + hh * 16 + d] * sK[s * 128 + hh * 16 + d];
            sc[s] = dot * 0.25f;
            mx = fmaxf(mx, sc[s]);
        }
        float sum = 0.f;
#pragma unroll
        for (int s = 0; s < 32; ++s) { sc[s] = __expf(sc[s] - mx); sum += sc[s]; }
        float inv = 1.0f / sum;
#pragma unroll
        for (int d = 0; d < 16; ++d) {
            float a = 0.f;
#pragma unroll
            for (int s = 0; s < 32; ++s) a += sc[s] * sV[s * 128 + hh * 16 + d];
            oacc[pp][d] = a * inv;
        }
    }
    __syncthreads();
    for (int pp = 0; pp < 2; ++pp) {  // o -> sZ (z dead)
        int p = tid + pp * 128;
        int t = p >> 3, hh = p & 7;
#pragma unroll
        for (int d = 0; d < 16; ++d) sZ[t * 128 + hh * 16 + d] = oacc[pp][d];
    }
    __syncthreads();

    // x_attend = o @ Wo + bo  (into registers, then sQ; q dead)
    float xa[32];
    for (int t = 0; t < 32; ++t) {
        float a = bo[tid];
        for (int f = 0; f < 128; ++f) a += sZ[t * 128 + f] * Wo[f * 128 + tid];
        xa[t] = a;
    }
    __syncthreads();
    for (int t = 0; t < 32; ++t) sQ[t * 128 + tid] = xa[t];
    __syncthreads();

    // MLP 128 -> 512 -> 128 in chunks of 128 hidden units (hidden chunk in sK)
    float mlpacc[32];
    for (int t = 0; t < 32; ++t) mlpacc[t] = 0.f;
    for (int c = 0; c < 4; ++c) {
        int i = c * 128 + tid;
        for (int t = 0; t < 32; ++t) {
            float a = bm1[i];
            for (int f = 0; f < 128; ++f) a += sQ[t * 128 + f] * Wm1[f * 512 + i];
            sK[t * 128 + tid] = fmaxf(a, 0.f);
        }
        __syncthreads();
        for (int t = 0; t < 32; ++t) {
            float a = 0.f;
            for (int ii = 0; ii < 128; ++ii)
                a += sK[t * 128 + ii] * Wm2[(c * 128 + ii) * 128 + tid];
            mlpacc[t] += a;
        }
        __syncthreads();
    }
    // residual
    for (int t = 0; t < 32; ++t) sQ[t * 128 + tid] = xa[t] + mlpacc[t];
    __syncthreads();

    // layernorm per row (threads 0..31 each own a row)
    if (tid < 32) {
        int t = tid;
        float mu = 0.f;
        for (int f = 0; f < 128; ++f) mu += sQ[t * 128 + f];
        mu *= (1.f / 128.f);
        float var = 0.f;
        for (int f = 0; f < 128; ++f) {
            float d = sQ[t * 128 + f] - mu;
            var += d * d;
        }
        var *= (1.f / 128.f);
        float inv = rsqrtf(var + 1e-5f);
        for (int f = 0; f < 128; ++f)
            sQ[t * 128 + f] = (sQ[t * 128 + f] - mu) * inv * g2[f] + beta2[f];
    }
    __syncthreads();

    // pooled = sum_t relu(x), then out = pooled @ Wl + bl
    float pooled = 0.f;
    for (int t = 0; t < 32; ++t) pooled += fmaxf(sQ[t * 128 + tid], 0.f);
    sK[tid] = pooled;
    __syncthreads();
    if (tid < 10) {
        float a = bl[tid];
        for (int j = 0; j < 128; ++j) a += sK[j] * Wl[j * 10 + tid];
        out[tid] = a;
    }
}

extern "C" void kernel_launch(void* const* d_in, const int* in_sizes, int n_in,
                              void* d_out, int out_size, void* d_ws, size_t ws_size,
                              hipStream_t stream) {
    const float* x  = (const float*)d_in[0];
    const int*   ei = (const int*)d_in[1];
    const float* W0 = (const float*)d_in[2];  const float* b0 = (const float*)d_in[3];
    const float* W1 = (const float*)d_in[4];  const float* b1 = (const float*)d_in[5];
    const float* W2 = (const float*)d_in[6];  const float* b2 = (const float*)d_in[7];
    const float* Wq = (const float*)d_in[8];  const float* bq = (const float*)d_in[9];
    const float* Wk = (const float*)d_in[10]; const float* bk = (const float*)d_in[11];
    const float* Wv = (const float*)d_in[12]; const float* bv = (const float*)d_in[13];
    const float* Wo = (const float*)d_in[14]; const float* bo = (const float*)d_in[15];
    const float* g2 = (const float*)d_in[16]; const float* beta2 = (const float*)d_in[17];
    const float* Wm1 = (const float*)d_in[18]; const float* bm1 = (const float*)d_in[19];
    const float* Wm2 = (const float*)d_in[20]; const float* bm2 = (const float*)d_in[21];
    const float* Wl  = (const float*)d_in[22]; const float* bl  = (const float*)d_in[23];
    float* out = (float*)d_out;

    float* ws   = (float*)d_ws;
    float* bufA = ws;               // [T,N,128] hW (scatter source)
    float* bufB = bufA + TNH_;      // [T,N,128] aggregation / h
    float* dinv = bufB + TNH_;      // [T,N]
    float* z    = dinv + TN_;       // [T,128]

    const int vecElems = TN_ * 32;                 // float4 groups per [T,N,128]
    const int vecBlocks = (vecElems + 255) / 256;  // 20000
    const int gemmBlocks = TN_ / 128;              // 1250

    // degree -> dinv (stored in dinv buffer)
    fill_deg<<<(TN_ + 255) / 256, 256, 0, stream>>>(dinv);
    accum_deg<<<(T_ * E_ + 255) / 256, 256, 0, stream>>>(ei, dinv);
    finalize_dinv<<<(TN_ + 255) / 256, 256, 0, stream>>>(dinv);

    // layer 0: x -> bufA (= x@W0) + bufB (= dinv^2 * x@W0) -> scatter -> tanh
    gemm_wmma<<<gemmBlocks, 256, 0, stream>>>(x, W0, dinv, bufA, bufB);
    edge_scatter<<<20000, 256, 0, stream>>>(bufA, ei, dinv, bufB);
    bias_tanh<<<vecBlocks, 256, 0, stream>>>(bufB, b0);

    // layer 1
    gemm_wmma<<<gemmBlocks, 256, 0, stream>>>(bufB, W1, dinv, bufA, bufB);
    edge_scatter<<<20000, 256, 0, stream>>>(bufA, ei, dinv, bufB);
    bias_tanh<<<vecBlocks, 256, 0, stream>>>(bufB, b1);

    // layer 2
    gemm_wmma<<<gemmBlocks, 256, 0, stream>>>(bufB, W2, dinv, bufA, bufB);
    edge_scatter<<<20000, 256, 0, stream>>>(bufA, ei, dinv, bufB);
    bias_tanh<<<vecBlocks, 256, 0, stream>>>(bufB, b2);

    // graph embeddings + head
    reduce_z<<<T_, 128, 0, stream>>>(bufB, z);
    head_kernel<<<1, 128, 0, stream>>>(z, Wq, bq, Wk, bk, Wv, bv, Wo, bo,
                                       g2, beta2, Wm1, bm1, Wm2, bm2, Wl, bl, out);
}